// Graphgnn_68453188764135
// MI455X (gfx1250) — compile-verified
//
#include <hip/hip_runtime.h>

// GraphConv x2 for MI455X (gfx1250, wave32, WMMA).
// out = relu(x@W1r^T + seg_sum(x[src],dst)@W1l^T + b1)  -> h
// out = relu(h@W2r^T + seg_sum(h[src],dst)@W2l^T + b2)  -> d_out
//
// D = 64 fixed. GEMMs use V_WMMA_F32_16X16X4_F32 (exact fp32).

typedef __attribute__((ext_vector_type(2))) float v2f;
typedef __attribute__((ext_vector_type(8))) float v8f;

#define DD 64

// ---------------------------------------------------------------- zero agg
__global__ void gnn_zero_kernel(float4* __restrict__ p, int n4) {
    int i = blockIdx.x * blockDim.x + threadIdx.x;
    if (i < n4) {
        float4 z; z.x = 0.f; z.y = 0.f; z.z = 0.f; z.w = 0.f;
        p[i] = z;
    }
}

// ------------------------------------------------- edge gather + scatter-add
// One thread per (edge, 4-float chunk): float4 gather from h[src], 4 native
// fp32 atomics into agg[dst]. All tables are L2-resident (25.6 MB << 192 MB).
__global__ __launch_bounds__(256)
void gnn_scatter_kernel(const float* __restrict__ h,
                        const int* __restrict__ src,
                        const int* __restrict__ dst,
                        float* __restrict__ agg, int nE) {
    int i = blockIdx.x * blockDim.x + threadIdx.x;
    int e = i >> 4;
    int q = i & 15;
    if (e >= nE) return;
    int s = src[e];
    int t = dst[e];
    const float4 v = *(const float4*)(h + s * DD + q * 4);
    float* b = agg + t * DD + q * 4;
#if defined(__HIP_DEVICE_COMPILE__)
    unsafeAtomicAdd(b + 0, v.x);   // -> global_atomic_add_f32 (no CAS loop)
    unsafeAtomicAdd(b + 1, v.y);
    unsafeAtomicAdd(b + 2, v.z);
    unsafeAtomicAdd(b + 3, v.w);
#else
    atomicAdd(b + 0, v.x); atomicAdd(b + 1, v.y);
    atomicAdd(b + 2, v.z); atomicAdd(b + 3, v.w);
#endif
}

// ----------------------------------------------------- fused dual-GEMM + relu
// OUT[16t : 16t+16, :] = relu(X_tile @ Wr^T + AGG_tile @ Wl^T + bias)
// One wave (32 lanes) per 16-row tile; 4 column tiles of 16 held in 4 v8f
// accumulators; K swept once in steps of 4 with V_WMMA_F32_16X16X4_F32.
//
// f32 16x16x4 operand layout (ISA 7.12.2):
//   A (16x4, MxK): lane&15 = M, lane>>4 selects K pair {0,1}/{2,3}, 2 VGPRs = K
//   B (4x16, KxN): lane&15 = N, lane>>4 selects K pair,            2 VGPRs = K
//   C/D (16x16) : lane&15 = N, VGPR j -> M = j + (lane>>4)*8
// B[k,n] = W[n*64 + k]  (W row-major, we need W^T) -> contiguous float2 in k.
__device__ __forceinline__ void gnn_store_tile(v8f acc,
                                               const float* __restrict__ bias,
                                               float* __restrict__ OUT,
                                               int rowbase, int nt, int lo, int hi) {
    float bv = bias[nt * 16 + lo];
#pragma unroll
    for (int j = 0; j < 8; ++j) {
        float v = acc[j] + bv;
        v = v > 0.f ? v : 0.f;
        OUT[(rowbase + hi * 8 + j) * DD + nt * 16 + lo] = v;
    }
}

__global__ __launch_bounds__(256)
void gnn_gemm_relu_kernel(const float* __restrict__ X,
                          const float* __restrict__ AGG,
                          const float* __restrict__ Wr,
                          const float* __restrict__ Wl,
                          const float* __restrict__ bias,
                          float* __restrict__ OUT,
                          int nTiles) {
    const int lane = threadIdx.x & 31;
    const int wave = threadIdx.x >> 5;
    const int tile = blockIdx.x * 8 + wave;
    if (tile >= nTiles) return;              // wave-uniform: EXEC all-1s inside

    const int lo = lane & 15;
    const int hi = lane >> 4;
    const int rowbase = tile * 16;

    const float* xrow = X   + (rowbase + lo) * DD;
    const float* grow = AGG + (rowbase + lo) * DD;
    const float* wr0 = Wr + (0 * 16 + lo) * DD;
    const float* wr1 = Wr + (1 * 16 + lo) * DD;
    const float* wr2 = Wr + (2 * 16 + lo) * DD;
    const float* wr3 = Wr + (3 * 16 + lo) * DD;
    const float* wl0 = Wl + (0 * 16 + lo) * DD;
    const float* wl1 = Wl + (1 * 16 + lo) * DD;
    const float* wl2 = Wl + (2 * 16 + lo) * DD;
    const float* wl3 = Wl + (3 * 16 + lo) * DD;

    v8f acc0 = {}, acc1 = {}, acc2 = {}, acc3 = {};

#pragma unroll 4
    for (int k = 0; k < DD; k += 4) {
        const int kk = k + hi * 2;
        v2f ax = *(const v2f*)(xrow + kk);
        v2f ag = *(const v2f*)(grow + kk);

        v2f br0 = *(const v2f*)(wr0 + kk);
        v2f br1 = *(const v2f*)(wr1 + kk);
        v2f br2 = *(const v2f*)(wr2 + kk);
        v2f br3 = *(const v2f*)(wr3 + kk);
        v2f bl0 = *(const v2f*)(wl0 + kk);
        v2f bl1 = *(const v2f*)(wl1 + kk);
        v2f bl2 = *(const v2f*)(wl2 + kk);
        v2f bl3 = *(const v2f*)(wl3 + kk);

        acc0 = __builtin_amdgcn_wmma_f32_16x16x4_f32(false, ax, false, br0, (short)0, acc0, false, false);
        acc1 = __builtin_amdgcn_wmma_f32_16x16x4_f32(false, ax, false, br1, (short)0, acc1, false, false);
        acc2 = __builtin_amdgcn_wmma_f32_16x16x4_f32(false, ax, false, br2, (short)0, acc2, false, false);
        acc3 = __builtin_amdgcn_wmma_f32_16x16x4_f32(false, ax, false, br3, (short)0, acc3, false, false);
        acc0 = __builtin_amdgcn_wmma_f32_16x16x4_f32(false, ag, false, bl0, (short)0, acc0, false, false);
        acc1 = __builtin_amdgcn_wmma_f32_16x16x4_f32(false, ag, false, bl1, (short)0, acc1, false, false);
        acc2 = __builtin_amdgcn_wmma_f32_16x16x4_f32(false, ag, false, bl2, (short)0, acc2, false, false);
        acc3 = __builtin_amdgcn_wmma_f32_16x16x4_f32(false, ag, false, bl3, (short)0, acc3, false, false);
    }

    gnn_store_tile(acc0, bias, OUT, rowbase, 0, lo, hi);
    gnn_store_tile(acc1, bias, OUT, rowbase, 1, lo, hi);
    gnn_store_tile(acc2, bias, OUT, rowbase, 2, lo, hi);
    gnn_store_tile(acc3, bias, OUT, rowbase, 3, lo, hi);
}

// ---------------------------------------------------------------- launcher
extern "C" void kernel_launch(void* const* d_in, const int* in_sizes, int n_in,
                              void* d_out, int out_size, void* d_ws, size_t ws_size,
                              hipStream_t stream) {
    const float* x      = (const float*)d_in[0];
    const int*   eidx   = (const int*)d_in[1];   // [2, E]
    // d_in[2] = dropout (0) — ignored
    const float* W1r    = (const float*)d_in[3];
    const float* W1l    = (const float*)d_in[4];
    const float* b1     = (const float*)d_in[5];
    const float* W2r    = (const float*)d_in[6];
    const float* W2l    = (const float*)d_in[7];
    const float* b2     = (const float*)d_in[8];
    float* out = (float*)d_out;

    const int N = in_sizes[0] / DD;          // 100000
    const int E = in_sizes[1] / 2;           // 1000000
    const int* src = eidx;
    const int* dst = eidx + E;

    float* agg = (float*)d_ws;               // [N, 64]
    float* h   = agg + (size_t)N * DD;       // [N, 64]

    const int n4        = N * (DD / 4);
    const int zeroGrid  = (n4 + 255) / 256;
    const int scatThr   = E * 16;
    const int scatGrid  = (scatThr + 255) / 256;
    const int nTiles    = N / 16;            // 6250 (N divisible by 16)
    const int gemmGrid  = (nTiles + 7) / 8;  // 8 waves / block

    // ---- layer 1
    gnn_zero_kernel<<<zeroGrid, 256, 0, stream>>>((float4*)agg, n4);
    gnn_scatter_kernel<<<scatGrid, 256, 0, stream>>>(x, src, dst, agg, E);
    gnn_gemm_relu_kernel<<<gemmGrid, 256, 0, stream>>>(x, agg, W1r, W1l, b1, h, nTiles);

    // ---- layer 2
    gnn_zero_kernel<<<zeroGrid, 256, 0, stream>>>((float4*)agg, n4);
    gnn_scatter_kernel<<<scatGrid, 256, 0, stream>>>(h, src, dst, agg, E);
    gnn_gemm_relu_kernel<<<gemmGrid, 256, 0, stream>>>(h, agg, W2r, W2l, b2, out, nTiles);
}